// DiffusionPolicy_3384434230026
// MI455X (gfx1250) — compile-verified
//
#include <hip/hip_runtime.h>
#include <hip/hip_bf16.h>

#define HID 128
#define VOCAB 20
#define KF 388            // final-MLP feature dim 387 padded to multiple of 4
#define AS_STRIDE 132     // LDS row stride (mod 64 banks == 4 -> conflict-free 16x4 frag reads)

typedef __attribute__((ext_vector_type(2))) float v2f;
typedef __attribute__((ext_vector_type(8))) float v8f;

// ---------------- elementwise helpers ----------------

__global__ void zero_f(float* __restrict__ p, int n) {
    int i = blockIdx.x * blockDim.x + threadIdx.x;
    if (i < n) p[i] = 0.0f;
}

__global__ void deg_init(float* __restrict__ deg, int n) {
    int i = blockIdx.x * blockDim.x + threadIdx.x;
    if (i < n) deg[i] = 1.0f;   // self-loop contributes 1
}

__global__ void deg_count(const int* __restrict__ ei, int E, float* __restrict__ deg) {
    int e = blockIdx.x * blockDim.x + threadIdx.x;
    if (e < E) unsafeAtomicAdd(&deg[ei[E + e]], 1.0f);   // dst = edge_index[1]
}

__global__ void deg_rsqrt(float* __restrict__ deg, int n) {
    int i = blockIdx.x * blockDim.x + threadIdx.x;
    if (i < n) deg[i] = rsqrtf(deg[i]);    // deg >= 1 always (self-loops)
}

// h[i][c] = W1[res[i]][c] + pos_i . W1[VOCAB..VOCAB+2][c]   (one-hot GEMM folded)
// fused: also writes self-loop-initialized accumulator agg[i][c] = h * dinv[i]^2
__global__ void featurize_fused(const int* __restrict__ res, const float* __restrict__ pos,
                                const float* __restrict__ W1, const float* __restrict__ dinv,
                                float* __restrict__ h, float* __restrict__ agg, int N) {
    int i = blockIdx.x * blockDim.x + threadIdx.x;
    if (i >= N * HID) return;
    int n = i >> 7, c = i & (HID - 1);
    float v = W1[res[n] * HID + c];
    v = fmaf(pos[n * 3 + 0], W1[(VOCAB + 0) * HID + c], v);
    v = fmaf(pos[n * 3 + 1], W1[(VOCAB + 1) * HID + c], v);
    v = fmaf(pos[n * 3 + 2], W1[(VOCAB + 2) * HID + c], v);
    float di = dinv[n];
    h[i] = v;
    agg[i] = v * di * di;
}

// one wave32 per edge: lane l moves channels [4l..4l+3] as float4 + 4 fp32 atomics
__global__ __launch_bounds__(256) void scatter_edges(const int* __restrict__ ei, int E,
                                                     const float* __restrict__ dinv,
                                                     const float* __restrict__ h,
                                                     float* __restrict__ agg) {
    int e = blockIdx.x * 8 + (threadIdx.x >> 5);
    if (e >= E) return;
    int lane = threadIdx.x & 31;
    int s = ei[e];
    int d = ei[E + e];
    float norm = dinv[s] * dinv[d];
    const float4* hv = (const float4*)(h + (size_t)s * HID);
    float4 v = hv[lane];
    float* base = agg + (size_t)d * HID + lane * 4;
    unsafeAtomicAdd(base + 0, v.x * norm);
    unsafeAtomicAdd(base + 1, v.y * norm);
    unsafeAtomicAdd(base + 2, v.z * norm);
    unsafeAtomicAdd(base + 3, v.w * norm);
}

__global__ void bias_relu(float* __restrict__ x, const float* __restrict__ b, int N) {
    int i = blockIdx.x * blockDim.x + threadIdx.x;
    if (i >= N * HID) return;
    x[i] = fmaxf(x[i] + b[i & (HID - 1)], 0.0f);
}

// ------- fused WMMA f32 GEMM: C = relu(Araw + b1) @ W ; aggOut = C * dinv^2 -------
// block = 256 threads = 8 waves; block handles a 16-row stripe; wave w does N-tile w*16.
// Fuses conv-1 epilogue (bias+relu on A-tile load) and conv-2 prologue (self-loop
// accumulator init on store). Uses V_WMMA_F32_16X16X4_F32 (f32, matches reference).
__global__ __launch_bounds__(256) void gemm128_wmma_fused(const float* __restrict__ Araw,
                                                          const float* __restrict__ b1,
                                                          const float* __restrict__ W,
                                                          const float* __restrict__ dinv,
                                                          float* __restrict__ C,
                                                          float* __restrict__ aggOut,
                                                          int M) {
    __shared__ float As[16 * AS_STRIDE];
    const int row0 = blockIdx.x * 16;
    const int wave = threadIdx.x >> 5;
    const int lane = threadIdx.x & 31;

    // cooperative A-tile load with fused bias+relu (clamped; no early return:
    // EXEC must stay all-1 for WMMA)
    for (int i = threadIdx.x; i < 16 * HID; i += 256) {
        int r = i >> 7, c = i & (HID - 1);
        int rr = min(row0 + r, M - 1);
        As[r * AS_STRIDE + c] = fmaxf(Araw[(size_t)rr * HID + c] + b1[c], 0.0f);
    }
    __syncthreads();

    const int n0    = wave * 16;
    const int mrow  = lane & 15;          // M (A) / N (B) index within tile
    const int khalf = (lane >> 4) * 2;    // lanes 16-31 hold K+2 / K+3

    v8f acc = {0.f, 0.f, 0.f, 0.f, 0.f, 0.f, 0.f, 0.f};
#pragma unroll
    for (int k = 0; k < HID; k += 4) {
        v2f a, b;
        // A 16x4: vgpr0 -> K = k+khalf, vgpr1 -> K = k+khalf+1
        a.x = As[mrow * AS_STRIDE + k + khalf];
        a.y = As[mrow * AS_STRIDE + k + khalf + 1];
        // B 4x16: same K striping, N = lane&15 across lanes within each vgpr
        b.x = W[(k + khalf) * HID + n0 + mrow];
        b.y = W[(k + khalf + 1) * HID + n0 + mrow];
        acc = __builtin_amdgcn_wmma_f32_16x16x4_f32(false, a, false, b,
                                                    (short)0, acc, false, false);
    }

    // C/D layout: vgpr v, lanes 0-15 -> M=v, lanes 16-31 -> M=v+8; N = lane&15
    const int nOut  = n0 + (lane & 15);
    const int mBase = row0 + (lane >> 4) * 8;
#pragma unroll
    for (int v = 0; v < 8; ++v) {
        int m = mBase + v;
        if (m < M) {
            float di = dinv[m];
            float val = acc[v];
            C[(size_t)m * HID + nOut] = val;
            aggOut[(size_t)m * HID + nOut] = val * di * di;   // self-loop init for conv-2
        }
    }
}

// ---------------- pooling ----------------

__global__ void pool_accum(const float* __restrict__ x, const int* __restrict__ batch,
                           float* __restrict__ sum, float* __restrict__ cnt, int N) {
    int i = blockIdx.x * blockDim.x + threadIdx.x;
    if (i >= N * HID) return;
    int n = i >> 7, c = i & (HID - 1);
    int b = batch[n];
    unsafeAtomicAdd(&sum[b * HID + c], x[i]);
    if (c == 0) unsafeAtomicAdd(&cnt[b], 1.0f);
}

__global__ void pool_div(float* __restrict__ sum, const float* __restrict__ cnt, int B) {
    int i = blockIdx.x * blockDim.x + threadIdx.x;
    if (i >= B * HID) return;
    sum[i] /= fmaxf(cnt[i >> 7], 1.0f);
}

// ---------------- timestep MLP: temb = relu(t*Wt1+bt1) @ Wt2 + bt2 ----------------
__global__ __launch_bounds__(HID) void t_mlp(const int* __restrict__ t,
                                             const float* __restrict__ Wt1, const float* __restrict__ bt1,
                                             const float* __restrict__ Wt2, const float* __restrict__ bt2,
                                             float* __restrict__ temb) {
    __shared__ float a1[HID];
    int c = threadIdx.x;
    float tf = (float)t[blockIdx.x];
    a1[c] = fmaxf(fmaf(tf, Wt1[c], bt1[c]), 0.0f);
    __syncthreads();
    float s = bt2[c];
    for (int k = 0; k < HID; ++k) s = fmaf(a1[k], Wt2[k * HID + c], s);
    temb[blockIdx.x * HID + c] = s;
}

// ---------------- final noise-prediction MLP, WMMA version ----------------
// 16 molecule nodes per block. feat [16 x 388] gathered to LDS (col 387 zero-pad),
// 8 waves produce hidden [16 x 128] via V_WMMA_F32_16X16X4_F32 (97 K-steps),
// then 48 threads do the tiny [128 x 3] projection.
__global__ __launch_bounds__(256) void final_mlp_wmma(const float* __restrict__ xm,
                                                      const float* __restrict__ pemb,
                                                      const float* __restrict__ temb,
                                                      const int* __restrict__ mbatch,
                                                      const float* __restrict__ na,
                                                      const float* __restrict__ Wn1,
                                                      const float* __restrict__ bn1,
                                                      const float* __restrict__ Wn2,
                                                      const float* __restrict__ bn2,
                                                      float* __restrict__ out, int NM) {
    __shared__ float feat[16 * KF];    // stride 388 (mod 64 == 4) -> conflict-free frag reads
    __shared__ float hid[16 * HID];
    const int node0 = blockIdx.x * 16;
    const int tid = threadIdx.x;

    // gather feature tile (clamped rows; no early return so EXEC stays all-1 for WMMA)
    for (int i = tid; i < 16 * KF; i += 256) {
        int r = i / KF, c = i - r * KF;
        int n = min(node0 + r, NM - 1);
        float v;
        if (c < HID)               v = xm[(size_t)n * HID + c];
        else if (c < 2 * HID)      v = pemb[mbatch[n] * HID + (c - HID)];
        else if (c < 3 * HID)      v = temb[mbatch[n] * HID + (c - 2 * HID)];
        else if (c < 3 * HID + 3)  v = na[n * 3 + (c - 3 * HID)];
        else                       v = 0.0f;   // zero pad K=387
        feat[i] = v;
    }
    __syncthreads();

    const int wave  = tid >> 5;
    const int lane  = tid & 31;
    const int n0    = wave * 16;           // hidden-channel tile
    const int mrow  = lane & 15;
    const int khalf = (lane >> 4) * 2;

    v8f acc = {0.f, 0.f, 0.f, 0.f, 0.f, 0.f, 0.f, 0.f};
#pragma unroll 4
    for (int k = 0; k < KF; k += 4) {
        v2f a, b;
        a.x = feat[mrow * KF + k + khalf];
        a.y = feat[mrow * KF + k + khalf + 1];
        int k0 = min(k + khalf, 3 * HID + 2);      // clamp: row 387 of Wn1 doesn't exist,
        int k1 = min(k + khalf + 1, 3 * HID + 2);  // matching A element is zero anyway
        b.x = Wn1[k0 * HID + n0 + mrow];
        b.y = Wn1[k1 * HID + n0 + mrow];
        acc = __builtin_amdgcn_wmma_f32_16x16x4_f32(false, a, false, b,
                                                    (short)0, acc, false, false);
    }

    // relu(hidden + bias) -> LDS
    const int nOut  = n0 + (lane & 15);
    const int mBase = (lane >> 4) * 8;
    float bias = bn1[nOut];
#pragma unroll
    for (int v = 0; v < 8; ++v)
        hid[(mBase + v) * HID + nOut] = fmaxf(acc[v] + bias, 0.0f);
    __syncthreads();

    // out[16 x 3] = hidden @ Wn2 + bn2
    if (tid < 48) {
        int r = tid / 3, j = tid - r * 3;
        int n = node0 + r;
        if (n < NM) {
            float o = bn2[j];
            for (int k = 0; k < HID; ++k) o = fmaf(hid[r * HID + k], Wn2[k * 3 + j], o);
            out[n * 3 + j] = o;
        }
    }
}

// ---------------- host-side orchestration ----------------

static inline int cdiv(int a, int b) { return (a + b - 1) / b; }

extern "C" void kernel_launch(void* const* d_in, const int* in_sizes, int n_in,
                              void* d_out, int out_size, void* d_ws, size_t ws_size,
                              hipStream_t stream) {
    const int*   pres = (const int*)  d_in[0];
    const float* ppos = (const float*)d_in[1];
    const int*   pei  = (const int*)  d_in[2];
    const int*   pb   = (const int*)  d_in[3];
    const int*   mres = (const int*)  d_in[4];
    const float* mpos = (const float*)d_in[5];
    const int*   mei  = (const int*)  d_in[6];
    const int*   mb   = (const int*)  d_in[7];
    const int*   tt   = (const int*)  d_in[8];
    const float* na   = (const float*)d_in[9];
    const float* Wp1  = (const float*)d_in[10]; const float* bp1 = (const float*)d_in[11];
    const float* Wp2  = (const float*)d_in[12]; const float* bp2 = (const float*)d_in[13];
    const float* Wm1  = (const float*)d_in[14]; const float* bm1 = (const float*)d_in[15];
    const float* Wm2  = (const float*)d_in[16]; const float* bm2 = (const float*)d_in[17];
    const float* Wt1  = (const float*)d_in[18]; const float* bt1 = (const float*)d_in[19];
    const float* Wt2  = (const float*)d_in[20]; const float* bt2 = (const float*)d_in[21];
    const float* Wn1  = (const float*)d_in[22]; const float* bn1 = (const float*)d_in[23];
    const float* Wn2  = (const float*)d_in[24]; const float* bn2 = (const float*)d_in[25];
    float* out = (float*)d_out;

    const int NP = in_sizes[0], EP = in_sizes[2] / 2;
    const int NM = in_sizes[4], EM = in_sizes[6] / 2;
    const int B  = in_sizes[8];

    float* ws = (float*)d_ws;
    float* degP = ws;  ws += NP;
    float* degM = ws;  ws += NM;
    float* pA   = ws;  ws += (size_t)NP * HID;
    float* pB   = ws;  ws += (size_t)NP * HID;
    float* mA   = ws;  ws += (size_t)NM * HID;
    float* mB   = ws;  ws += (size_t)NM * HID;
    float* psum = ws;  ws += (size_t)B * HID;
    float* pcnt = ws;  ws += B;
    float* temb = ws;  ws += (size_t)B * HID;

    const int T = 256;

    // ---- degree / normalization ----
    deg_init<<<cdiv(NP, T), T, 0, stream>>>(degP, NP);
    deg_init<<<cdiv(NM, T), T, 0, stream>>>(degM, NM);
    deg_count<<<cdiv(EP, T), T, 0, stream>>>(pei, EP, degP);
    deg_count<<<cdiv(EM, T), T, 0, stream>>>(mei, EM, degM);
    deg_rsqrt<<<cdiv(NP, T), T, 0, stream>>>(degP, NP);
    deg_rsqrt<<<cdiv(NM, T), T, 0, stream>>>(degM, NM);

    // ---- protein branch ----
    featurize_fused<<<cdiv(NP * HID, T), T, 0, stream>>>(pres, ppos, Wp1, degP, pA, pB, NP);
    scatter_edges<<<cdiv(EP, 8), T, 0, stream>>>(pei, EP, degP, pA, pB);
    gemm128_wmma_fused<<<cdiv(NP, 16), T, 0, stream>>>(pB, bp1, Wp2, degP, pA, pB, NP);
    scatter_edges<<<cdiv(EP, 8), T, 0, stream>>>(pei, EP, degP, pA, pB);
    bias_relu<<<cdiv(NP * HID, T), T, 0, stream>>>(pB, bp2, NP);

    // ---- molecule branch ----
    featurize_fused<<<cdiv(NM * HID, T), T, 0, stream>>>(mres, mpos, Wm1, degM, mA, mB, NM);
    scatter_edges<<<cdiv(EM, 8), T, 0, stream>>>(mei, EM, degM, mA, mB);
    gemm128_wmma_fused<<<cdiv(NM, 16), T, 0, stream>>>(mB, bm1, Wm2, degM, mA, mB, NM);
    scatter_edges<<<cdiv(EM, 8), T, 0, stream>>>(mei, EM, degM, mA, mB);
    bias_relu<<<cdiv(NM * HID, T), T, 0, stream>>>(mB, bm2, NM);

    // ---- protein mean pool ----
    zero_f<<<cdiv(B * (HID + 1), T), T, 0, stream>>>(psum, B * (HID + 1)); // psum + pcnt contiguous
    pool_accum<<<cdiv(NP * HID, T), T, 0, stream>>>(pB, pb, psum, pcnt, NP);
    pool_div<<<cdiv(B * HID, T), T, 0, stream>>>(psum, pcnt, B);

    // ---- timestep MLP ----
    t_mlp<<<B, HID, 0, stream>>>(tt, Wt1, bt1, Wt2, bt2, temb);

    // ---- final noise predictor (WMMA) ----
    final_mlp_wmma<<<cdiv(NM, 16), T, 0, stream>>>(mB, psum, temb, mb, na,
                                                   Wn1, bn1, Wn2, bn2, out, NM);
}